// ManyToOneRNN_27547920236739
// MI455X (gfx1250) — compile-verified
//
#include <hip/hip_runtime.h>

typedef float v2f __attribute__((ext_vector_type(2)));
typedef float v8f __attribute__((ext_vector_type(8)));

// lane L <-> lane L^16 exchange. Prefer v_permlanex16_b32 (VALU cross-lane,
// stays off the LDS pipe / DScnt); identity nibble selects swap the halves.
__device__ __forceinline__ float xor16(float v) {
#if __has_builtin(__builtin_amdgcn_permlanex16)
  int i = __builtin_bit_cast(int, v);
  int r = __builtin_amdgcn_permlanex16(i, i, 0x76543210u, 0xfedcba98u,
                                       true /*fi*/, false /*bound_ctrl*/);
  return __builtin_bit_cast(float, r);
#else
  return __shfl_xor(v, 16, 32);
#endif
}

__device__ __forceinline__ float fast_tanh(float x) {
#if __has_builtin(__builtin_amdgcn_tanhf)
  return __builtin_amdgcn_tanhf(x);        // single v_tanh_f32 TRANS op
#else
  float e = __builtin_amdgcn_exp2f(x * 2.88539008177792681472f); // 2*log2(e)
  return 1.0f - 2.0f * __builtin_amdgcn_rcpf(e + 1.0f);
#endif
}

__device__ __forceinline__ float fast_sigmoid(float x) {
  float e = __builtin_amdgcn_exp2f(x * -1.44269504088896340736f); // -log2(e)
  return __builtin_amdgcn_rcpf(1.0f + e);
}

#define WMMA4(A, Bv, C) \
  __builtin_amdgcn_wmma_f32_16x16x4_f32(false, (A), false, (Bv), (short)0, (C), false, false)

// One wave owns 16 batch rows for all T steps. State kept transposed:
// G = h^T (16 hidden x 16 batch), so A operands (W_hh, W_ih) are loop
// invariant; only the B operand (G) changes per step.
__global__ __launch_bounds__(256) void rnn_wmma_kernel(
    const float* __restrict__ x,
    const float* __restrict__ Wih, const float* __restrict__ Whh,
    const float* __restrict__ bih, const float* __restrict__ bhh,
    const float* __restrict__ Wfc, const float* __restrict__ bfc,
    float* __restrict__ out, int B, int T)
{
  const int wave = (int)((blockIdx.x * blockDim.x + threadIdx.x) >> 5);
  const int lane = (int)(threadIdx.x & 31u);
  const int bb = wave * 16;
  if (bb >= B) return;                 // wave-uniform: EXEC stays all-ones

  const int hi  = lane >> 4;           // half-wave select
  const int col = lane & 15;           // batch column (and A-matrix row)

  int row = bb + col; if (row >= B) row = B - 1;   // clamp for ragged tail
  const float* __restrict__ xrow = x + (size_t)row * (size_t)T * 5u;

  // --- loop-invariant A operands (16x4 f32 A-layout: lane m holds K=2*hi,2*hi+1)
  v2f aih0, aih1;
  aih0.x = Wih[col * 5 + 2 * hi];
  aih0.y = Wih[col * 5 + 2 * hi + 1];
  aih1.x = hi ? 0.0f : Wih[col * 5 + 4];   // K=4 only; K=5..7 zero-padded in A
  aih1.y = 0.0f;

  v2f ahh0, ahh1, ahh2, ahh3;              // W_hh split over K=16 in 4 chunks
  ahh0.x = Whh[col*16 +  0 + 2*hi]; ahh0.y = Whh[col*16 +  1 + 2*hi];
  ahh1.x = Whh[col*16 +  4 + 2*hi]; ahh1.y = Whh[col*16 +  5 + 2*hi];
  ahh2.x = Whh[col*16 +  8 + 2*hi]; ahh2.y = Whh[col*16 +  9 + 2*hi];
  ahh3.x = Whh[col*16 + 12 + 2*hi]; ahh3.y = Whh[col*16 + 13 + 2*hi];

  v8f cbv;          // (b_ih + b_hh) broadcast into C/D layout (rows = hidden)
  float wfc[8];
  #pragma unroll
  for (int v = 0; v < 8; ++v) {
    cbv[v] = bih[v + 8*hi] + bhh[v + 8*hi];
    wfc[v] = Wfc[v + 8*hi];
  }
  const float bfc0 = bfc[0];

  const v8f vzero = {};

  float g[8];                                // G = h^T in C/D layout; h0 = 0
  #pragma unroll
  for (int v = 0; v < 8; ++v) g[v] = 0.0f;

  const float* p = xrow;                     // advances 5 floats (20 B) / step

  // One timestep. Loads have no loop-carried dependence -> compiler may hoist
  // them ahead; the g-chain is: permlanex16 -> cndmask -> 1 WMMA (4 issued
  // independently, C=proj/0) -> 2 add levels -> v_tanh.
  auto step = [&](bool prefetch) {
    const float nx0 = p[2*hi];
    const float nx1 = p[2*hi + 1];
    const float nx4 = p[4];
    if (prefetch) __builtin_prefetch(p + 320);   // 64 steps (10 lines) ahead

    // input projection (independent of g): m = Wih @ x_t^T + bias
    v2f bx0; bx0.x = nx0; bx0.y = nx1;
    v2f bx1; bx1.x = nx4; bx1.y = 0.0f;   // hi-lane garbage rows killed by aih1==0
    v8f m = WMMA4(aih0, bx0, cbv);
    m = WMMA4(aih1, bx1, m);

    // D-layout -> four K=4 B-tiles: 8 half-swaps + selects
    float sg[8];
    #pragma unroll
    for (int v = 0; v < 8; ++v) sg[v] = xor16(g[v]);

    v2f b0; b0.x = hi ? sg[2] : g[0]; b0.y = hi ? sg[3] : g[1];   // K=0..3
    v2f b1; b1.x = hi ? sg[6] : g[4]; b1.y = hi ? sg[7] : g[5];   // K=4..7
    v2f b2; b2.x = hi ? g[2] : sg[0]; b2.y = hi ? g[3] : sg[1];   // K=8..11
    v2f b3; b3.x = hi ? g[6] : sg[4]; b3.y = hi ? g[7] : sg[5];   // K=12..15

    // 4 independent WMMAs (pipelined back-to-back), then a 2-level add tree
    v8f r0 = WMMA4(ahh0, b0, m);
    v8f r1 = WMMA4(ahh1, b1, vzero);
    v8f r2 = WMMA4(ahh2, b2, vzero);
    v8f r3 = WMMA4(ahh3, b3, vzero);
    v8f cur = (r0 + r1) + (r2 + r3);

    #pragma unroll
    for (int v = 0; v < 8; ++v) g[v] = fast_tanh(cur[v]);
    p += 5;
  };

  int t = 0;
  const int Tmain = (T > 64) ? (T - 64) : 0;
  for (; t < Tmain; ++t) step(true);   // with lookahead prefetch
  for (; t < T; ++t) step(false);      // tail: no prefetch past the buffer

  // fc + sigmoid: per-lane partial over this half's 8 hidden rows, xor16 reduce
  float pa = 0.0f;
  #pragma unroll
  for (int v = 0; v < 8; ++v) pa += g[v] * wfc[v];
  const float tot = pa + xor16(pa) + bfc0;
  const float s = fast_sigmoid(tot);
  const int orow = bb + col;
  if (!hi && orow < B) out[orow] = s;
}

extern "C" void kernel_launch(void* const* d_in, const int* in_sizes, int n_in,
                              void* d_out, int out_size, void* d_ws, size_t ws_size,
                              hipStream_t stream) {
  const float* x   = (const float*)d_in[0];
  const float* Wih = (const float*)d_in[1];
  const float* Whh = (const float*)d_in[2];
  const float* bih = (const float*)d_in[3];
  const float* bhh = (const float*)d_in[4];
  const float* Wfc = (const float*)d_in[5];
  const float* bfc = (const float*)d_in[6];
  float* out = (float*)d_out;

  const int B = out_size;                       // [B,1] output
  const int T = in_sizes[0] / (B * 5);          // x is [B,T,5]

  const int tiles = (B + 15) / 16;              // 16 batch rows per wave
  const int wavesPerBlock = 8;                  // 256 threads
  const int blocks = (tiles + wavesPerBlock - 1) / wavesPerBlock;

  rnn_wmma_kernel<<<blocks, wavesPerBlock * 32, 0, stream>>>(
      x, Wih, Whh, bih, bhh, Wfc, bfc, out, B, T);
}